// FLAGatedDeltaNet_24129126269510
// MI455X (gfx1250) — compile-verified
//
#include <hip/hip_runtime.h>
#include <math.h>

#define B_ 2
#define L_ 1024
#define D_ 1024
#define H_ 16
#define HD_ 64
#define M_ (B_*L_)   /* 2048 rows total */
#define EPS_ 1e-12f

typedef __attribute__((ext_vector_type(2))) float v2f;
typedef __attribute__((ext_vector_type(4))) float v4f;
typedef __attribute__((ext_vector_type(8))) float v8f;
typedef __attribute__((ext_vector_type(4))) unsigned int u32x4;
typedef __attribute__((ext_vector_type(8))) int i32x8;
typedef __attribute__((ext_vector_type(4))) int i32x4;

// Exact-fp32 WMMA: D = A(16x4) * B(4x16) + C(16x16)
__device__ __forceinline__ v8f wmma4(v2f a, v2f b, v8f c) {
  return __builtin_amdgcn_wmma_f32_16x16x4_f32(
      /*neg_a=*/false, a, /*neg_b=*/false, b,
      /*c_mod=*/(short)0, c, /*reuse_a=*/false, /*reuse_b=*/false);
}

// ---------------------------------------------------------------------------
// TDM: DMA a rows x cols fp32 tile (row stride in elements) from global
// memory into LDS.  Descriptor per CDNA5 ISA ch.8 (D# groups 0/1), 2D tensor,
// remaining groups zero (unused for 2D).  Tracked by TENSORcnt.
// 6-arg builtin form (clang-23 / therock headers).
// ---------------------------------------------------------------------------
__device__ __forceinline__ void tdm_load_tile_f32(const float* gsrc,
                                                  float* lds_dst,
                                                  int rows, int cols,
                                                  int row_stride_elems) {
  unsigned long long ga = (unsigned long long)(const void*)gsrc;
  unsigned lds = (unsigned)(unsigned long long)(void*)lds_dst; // LDS byte offset
  u32x4 g0;
  g0[0] = 1u;                                   // count=1 (valid), user mode
  g0[1] = lds;                                  // lds_addr (bytes)
  g0[2] = (unsigned)ga;                         // global_addr[31:0]
  g0[3] = (unsigned)((ga >> 32) & 0x01FFFFFFu)  // global_addr[56:32]
          | (2u << 30);                         // type = 2 ("image")
  i32x8 g1;
  g1[0] = 0x00020000;                           // data_size=2 -> 4 bytes; mask=0
  g1[1] = (int)(((unsigned)cols & 0xFFFFu) << 16);          // tensor_dim0[15:0]
  g1[2] = (int)((((unsigned)cols >> 16) & 0xFFFFu)          // tensor_dim0[31:16]
          | (((unsigned)rows & 0xFFFFu) << 16));            // tensor_dim1[15:0]
  g1[3] = (int)((((unsigned)rows >> 16) & 0xFFFFu)          // tensor_dim1[31:16]
          | (((unsigned)cols & 0xFFFFu) << 16));            // tile_dim0
  g1[4] = (int)((unsigned)rows & 0xFFFFu);                  // tile_dim1 (tile_dim2=0)
  g1[5] = row_stride_elems;                     // tensor_dim0_stride[31:0]
  g1[6] = 0;                                    // stride[47:32] | dim1_stride[15:0]
  g1[7] = 0;
  i32x4 z4 = {0, 0, 0, 0};
  i32x8 z8 = {0, 0, 0, 0, 0, 0, 0, 0};
  __builtin_amdgcn_tensor_load_to_lds(g0, g1, z4, z4, z8, 0);
}

// ---------------------------------------------------------------------------
// beta = sigmoid(x @ Wb + bb)   [M, H]
// ---------------------------------------------------------------------------
__global__ void beta_kernel(const float* __restrict__ x,
                            const float* __restrict__ Wb,
                            const float* __restrict__ bbias,
                            float* __restrict__ beta) {
  int idx = blockIdx.x * blockDim.x + threadIdx.x;  // 0 .. M*H-1
  int row = idx >> 4;
  int h   = idx & 15;
  float s = bbias[h];
  const float* xr = x + (size_t)row * D_;
  for (int d = 0; d < D_; ++d) s += xr[d] * Wb[d * H_ + h];
  beta[idx] = 1.0f / (1.0f + expf(-s));
}

// ---------------------------------------------------------------------------
// q,k,v = x @ {Wq,Wk,Wv}.  Block = 32 rows x 64 cols (one head), 8 waves.
// Weight panels (64x64 per K-slice) are DMA'd into LDS by the TDM; waves
// read B operands from LDS and A (x) as contiguous float2 from global.
// ---------------------------------------------------------------------------
__global__ void __launch_bounds__(256)
qkv_kernel(const float* __restrict__ x,
           const float* __restrict__ Wq,
           const float* __restrict__ Wk,
           const float* __restrict__ Wv,
           float* __restrict__ q, float* __restrict__ k, float* __restrict__ v) {
  __shared__ float sW[3][64 * 64];          // 48 KB weight panels
  const int lane = threadIdx.x & 31;
  const int w    = threadIdx.x >> 5;        // 0..7
  const int n    = lane & 15;
  const int g    = lane >> 4;
  const int rt   = w & 1;                   // row tile 0..1
  const int ct   = w >> 1;                  // col tile 0..3
  const int row0 = blockIdx.x * 32 + rt * 16;
  const int col0 = blockIdx.y * 64;         // 64-wide head panel

  v8f aq = {}, ak = {}, av = {};
  const float* xrow = x + (size_t)(row0 + n) * D_;

  for (int ks = 0; ks < 16; ++ks) {
    __syncthreads();                        // panel consumers from prev slice done
    if (w == 0) {
      const size_t wof = (size_t)(ks * 64) * D_ + col0;
      tdm_load_tile_f32(Wq + wof, sW[0], 64, 64, D_);
      tdm_load_tile_f32(Wk + wof, sW[1], 64, 64, D_);
      tdm_load_tile_f32(Wv + wof, sW[2], 64, 64, D_);
      __builtin_amdgcn_s_wait_tensorcnt(0);
    }
    __syncthreads();                        // panels visible to all waves

    const float* xs = xrow + ks * 64;
    for (int kk = 0; kk < 64; kk += 4) {
      v2f a = *(const v2f*)(xs + kk + 2 * g);
      int bi = (kk + 2 * g) * 64 + ct * 16 + n;
      v2f bq, bk, bv;
      bq.x = sW[0][bi]; bq.y = sW[0][bi + 64];
      bk.x = sW[1][bi]; bk.y = sW[1][bi + 64];
      bv.x = sW[2][bi]; bv.y = sW[2][bi + 64];
      aq = wmma4(a, bq, aq);
      ak = wmma4(a, bk, ak);
      av = wmma4(a, bv, av);
    }
  }
  for (int r = 0; r < 8; ++r) {
    size_t o = (size_t)(row0 + r + 8 * g) * D_ + col0 + ct * 16 + n;
    q[o] = aq[r]; k[o] = ak[r]; v[o] = av[r];
  }
}

// ---------------------------------------------------------------------------
// Per-(row,head) L2 normalization of q and k. One wave32 per 64-vector.
// ---------------------------------------------------------------------------
__global__ void norm_kernel(float* __restrict__ q, float* __restrict__ k) {
  int wid  = (blockIdx.x * blockDim.x + threadIdx.x) >> 5;  // 0 .. M*H-1
  int lane = threadIdx.x & 31;
  int row  = wid >> 4;
  int h    = wid & 15;
  size_t base = (size_t)row * D_ + h * HD_ + 2 * lane;
  v2f qv = *(v2f*)(q + base);
  v2f kv = *(v2f*)(k + base);
  float sq = qv.x * qv.x + qv.y * qv.y;
  float sk = kv.x * kv.x + kv.y * kv.y;
  for (int o = 16; o > 0; o >>= 1) {
    sq += __shfl_xor(sq, o, 32);
    sk += __shfl_xor(sk, o, 32);
  }
  float iq = 1.0f / fmaxf(sqrtf(sq), EPS_);
  float ik = 1.0f / fmaxf(sqrtf(sk), EPS_);
  qv.x *= iq; qv.y *= iq;
  kv.x *= ik; kv.y *= ik;
  *(v2f*)(q + base) = qv;
  *(v2f*)(k + base) = kv;
}

// ---------------------------------------------------------------------------
// Chunked causal linear attention. One block per (batch, head), 8 waves.
// Per 64-chunk:  P = tril(Qc @ (beta*K)cT);  O = P@Vc + Qc@S;  S += (beta*K)cT@Vc.
// ---------------------------------------------------------------------------
__global__ void __launch_bounds__(256)
attn_kernel(const float* __restrict__ q, const float* __restrict__ k,
            const float* __restrict__ v, const float* __restrict__ beta,
            float* __restrict__ o) {
  __shared__ float sK[64 * 64];   // beta-scaled K chunk
  __shared__ float sV[64 * 64];
  __shared__ float sP[64 * 64];   // masked score matrix
  __shared__ float sS[64 * 64];   // running state (prev chunks)

  const int bh    = blockIdx.x;        // 0..31
  const int batch = bh >> 4;
  const int h     = bh & 15;
  const int tid   = threadIdx.x;
  const int lane  = tid & 31;
  const int w     = tid >> 5;
  const int n     = lane & 15;
  const int g     = lane >> 4;

  const size_t rowbase = (size_t)batch * L_;
  const float* Qg = q + rowbase * D_ + h * HD_;
  const float* Kg = k + rowbase * D_ + h * HD_;
  const float* Vg = v + rowbase * D_ + h * HD_;
  float*       Og = o + rowbase * D_ + h * HD_;
  const float* Bg = beta + rowbase * H_ + h;

  v8f accS[2] = { {}, {} };

  for (int c = 0; c < 16; ++c) {
    const int rb = c * 64;

    // --- stage beta*K and V into LDS (16 floats / thread each) ---
    for (int rep = 0; rep < 4; ++rep) {
      int f  = rep * 1024 + tid * 4;   // flat index, row-major 64x64
      int rr = f >> 6, cc = f & 63;
      v4f vv = *(const v4f*)(Vg + (size_t)(rb + rr) * D_ + cc);
      *(v4f*)(sV + f) = vv;
      float bt = Bg[(size_t)(rb + rr) * H_];
      v4f kk4 = *(const v4f*)(Kg + (size_t)(rb + rr) * D_ + cc);
      kk4.x *= bt; kk4.y *= bt; kk4.z *= bt; kk4.w *= bt;
      *(v4f*)(sK + f) = kk4;
    }
    __syncthreads();

    // --- step 2: P = tril(Qc @ sK^T) ---
    for (int j = 0; j < 2; ++j) {
      int T = 2 * w + j, ti = T >> 2, tj = T & 3;
      v8f acc = {};
      if (tj <= ti) {
        const float* qrow = Qg + (size_t)(rb + ti * 16 + n) * D_;
        const float* krow = sK + (tj * 16 + n) * 64;
        for (int kk = 0; kk < 64; kk += 4) {
          v2f a = *(const v2f*)(qrow + kk + 2 * g);
          v2f b = *(const v2f*)(krow + kk + 2 * g);
          acc = wmma4(a, b, acc);
        }
        if (ti == tj) {                       // causal mask on diagonal tile
          for (int r = 0; r < 8; ++r)
            if (n > r + 8 * g) acc[r] = 0.0f;
        }
      }
      for (int r = 0; r < 8; ++r)
        sP[(ti * 16 + r + 8 * g) * 64 + tj * 16 + n] = acc[r];
    }
    __syncthreads();

    // --- step 3: O = P @ V  (+ Qc @ S_prev for c>0) ---
    for (int j = 0; j < 2; ++j) {
      int T = 2 * w + j, ti = T >> 2, tj = T & 3;
      v8f acc = {};
      const float* prow = sP + (ti * 16 + n) * 64;
      for (int kk = 0; kk < 64; kk += 4) {
        v2f a = *(const v2f*)(prow + kk + 2 * g);
        v2f b;
        b.x = sV[(kk + 2 * g) * 64 + tj * 16 + n];
        b.y = sV[(kk + 2 * g + 1) * 64 + tj * 16 + n];
        acc = wmma4(a, b, acc);
      }
      if (c > 0) {
        const float* qrow = Qg + (size_t)(rb + ti * 16 + n) * D_;
        for (int kk = 0; kk < 64; kk += 4) {
          v2f a = *(const v2f*)(qrow + kk + 2 * g);
          v2f b;
          b.x = sS[(kk + 2 * g) * 64 + tj * 16 + n];
          b.y = sS[(kk + 2 * g + 1) * 64 + tj * 16 + n];
          acc = wmma4(a, b, acc);
        }
      }
      for (int r = 0; r < 8; ++r)
        Og[(size_t)(rb + ti * 16 + r + 8 * g) * D_ + tj * 16 + n] = acc[r];
    }

    // --- step 4: S += sK^T @ V (register accumulators, 2 tiles/wave) ---
    for (int j = 0; j < 2; ++j) {
      int T = 2 * w + j, ti = T >> 2, tj = T & 3;
      for (int kk = 0; kk < 64; kk += 4) {
        v2f a, b;
        a.x = sK[(kk + 2 * g) * 64 + ti * 16 + n];
        a.y = sK[(kk + 2 * g + 1) * 64 + ti * 16 + n];
        b.x = sV[(kk + 2 * g) * 64 + tj * 16 + n];
        b.y = sV[(kk + 2 * g + 1) * 64 + tj * 16 + n];
        accS[j] = wmma4(a, b, accS[j]);
      }
    }
    __syncthreads();   // everyone done reading sS / sK / sV

    // --- step 5: publish S for next chunk ---
    for (int j = 0; j < 2; ++j) {
      int T = 2 * w + j, ti = T >> 2, tj = T & 3;
      for (int r = 0; r < 8; ++r)
        sS[(ti * 16 + r + 8 * g) * 64 + tj * 16 + n] = accS[j][r];
    }
    __syncthreads();
  }
}

// ---------------------------------------------------------------------------
// out = attn @ Wo.  Block = 64x64 tile, 8 waves x 2 sub-tiles.
// Wo panels staged to LDS by the TDM, one 64x64 slice per K-step group.
// ---------------------------------------------------------------------------
__global__ void __launch_bounds__(256)
out_kernel(const float* __restrict__ a_, const float* __restrict__ Wo,
           float* __restrict__ out) {
  __shared__ float sWo[64 * 64];            // 16 KB weight panel
  const int lane = threadIdx.x & 31;
  const int w    = threadIdx.x >> 5;
  const int n    = lane & 15;
  const int g    = lane >> 4;
  const int row0 = blockIdx.x * 64;
  const int col0 = blockIdx.y * 64;

  v8f acc[2] = { {}, {} };
  for (int ks = 0; ks < 16; ++ks) {
    __syncthreads();
    if (w == 0) {
      tdm_load_tile_f32(Wo + (size_t)(ks * 64) * D_ + col0, sWo, 64, 64, D_);
      __builtin_amdgcn_s_wait_tensorcnt(0);
    }
    __syncthreads();

    for (int j = 0; j < 2; ++j) {
      int T = 2 * w + j, ti = T >> 2, tj = T & 3;
      const float* arow = a_ + (size_t)(row0 + ti * 16 + n) * D_ + ks * 64;
      for (int kk = 0; kk < 64; kk += 4) {
        v2f a = *(const v2f*)(arow + kk + 2 * g);
        int bi = (kk + 2 * g) * 64 + tj * 16 + n;
        v2f b;
        b.x = sWo[bi];
        b.y = sWo[bi + 64];
        acc[j] = wmma4(a, b, acc[j]);
      }
    }
  }
  for (int j = 0; j < 2; ++j) {
    int T = 2 * w + j, ti = T >> 2, tj = T & 3;
    for (int r = 0; r < 8; ++r)
      out[(size_t)(row0 + ti * 16 + r + 8 * g) * D_ + col0 + tj * 16 + n] = acc[j][r];
  }
}

// ---------------------------------------------------------------------------
extern "C" void kernel_launch(void* const* d_in, const int* in_sizes, int n_in,
                              void* d_out, int out_size, void* d_ws, size_t ws_size,
                              hipStream_t stream) {
  const float* x     = (const float*)d_in[0];
  const float* Wq    = (const float*)d_in[1];
  const float* Wk    = (const float*)d_in[2];
  const float* Wv    = (const float*)d_in[3];
  const float* Wo    = (const float*)d_in[4];
  const float* Wb    = (const float*)d_in[5];
  const float* bbias = (const float*)d_in[6];

  float* ws   = (float*)d_ws;
  float* q    = ws;
  float* k    = q + (size_t)M_ * D_;
  float* v    = k + (size_t)M_ * D_;
  float* attn = v + (size_t)M_ * D_;
  float* beta = attn + (size_t)M_ * D_;   // M_*H_ floats

  beta_kernel<<<(M_ * H_) / 256, 256, 0, stream>>>(x, Wb, bbias, beta);

  dim3 g1(M_ / 32, H_);
  qkv_kernel<<<g1, 256, 0, stream>>>(x, Wq, Wk, Wv, q, k, v);

  norm_kernel<<<(M_ * H_ * 32) / 256, 256, 0, stream>>>(q, k);

  attn_kernel<<<B_ * H_, 256, 0, stream>>>(q, k, v, beta, attn);

  dim3 g4(M_ / 64, D_ / 64);
  out_kernel<<<g4, 256, 0, stream>>>(attn, Wo, (float*)d_out);
}